// CRFLoss_19670950216522
// MI455X (gfx1250) — compile-verified
//
#include <hip/hip_runtime.h>
#include <math.h>

// CRF forward (linear-chain NLL) for B=32, S=128, V=8, T=66.
// 16 single-wave (wave32) blocks, each owning 16 chains.
//
// Latency-oriented design: the forward recursion runs in the EXP domain:
//     p'_j = ( sum_i p_i * expT_ij ) * exp(emit_j),   renormalized per chain
//     logscale -= log(rcp(max_j p'_j))
// so the per-step serial chain is GEMM -> mul -> max -> rcp -> mul -> LDS,
// while exp(emissions) and log(scale) run in parallel with the GEMM.
// The 16x68 * 68x80 f32 GEMM uses V_WMMA_F32_16X16X4_F32 (17 K x 5 N tiles)
// with 2-way split accumulators (WMMA RAW depth 9 instead of 17).

typedef __attribute__((ext_vector_type(2))) float v2f;
typedef __attribute__((ext_vector_type(8))) float v8f;

#define BB   32
#define SS   128
#define VV   8
#define TT   66
#define TP   80    // padded N (5 tiles of 16)
#define KP   68    // padded K (17 tiles of 4)
#define NTI  5
#define NKI  17
#define NEG_BIG (-1.0e30f)

__global__ __launch_bounds__(32) void crf_zero(float* out) {
    if (threadIdx.x == 0) out[0] = 0.0f;
}

__global__ __launch_bounds__(32) void crf_fwd(
    const float* __restrict__ score,   // [B,S,S,T]
    const float* __restrict__ trans,   // [T,T]
    const float* __restrict__ startT,  // [T]
    const float* __restrict__ endT,    // [T]
    const int*   __restrict__ vlab,    // [B,V]
    const int*   __restrict__ rlab,    // [B,V,S]
    float*       __restrict__ out)     // [1]
{
    __shared__ float sExpT[KP * TP];   // exp(transitions), zero padded (21.8 KB)
    __shared__ float sP[16 * TP];      // normalized p matrix (5.1 KB)
    __shared__ float sRed[32];

    const int lane = threadIdx.x;      // single wave32
    const int hi   = lane >> 4;        // half of wave
    const int lh   = lane & 15;
    const int blk  = blockIdx.x;       // 16 blocks x 16 chains

    // ---- build exp(transitions) in LDS, zero padded (uniform trip count) ----
    for (int idx = lane; idx < KP * TP; idx += 32) {
        const int i = idx / TP, j = idx % TP;
        sExpT[idx] = (i < TT && j < TT) ? __expf(trans[i * TT + j]) : 0.0f;
    }

    // ---- gold (numerator) score: chain lh, halves split the sequence ----
    {
        const int bv  = blk * 16 + lh;
        const int b2  = bv >> 3, v2 = bv & 7;
        const int vl  = vlab[b2 * VV + v2];
        const int* tg = rlab + bv * SS;
        const float* em = score + (size_t)((b2 * SS + vl) * SS) * TT;
        float part = 0.0f;
        const int s0 = hi * 64, s1 = s0 + 64;
        for (int s = s0; s < s1; ++s) part += em[s * TT + tg[s]];
        const int p0 = hi ? 64 : 0, p1 = hi ? 127 : 64;
        for (int s = p0; s < p1; ++s) part += trans[tg[s] * TT + tg[s + 1]];
        part += hi ? endT[tg[SS - 1]] : startT[tg[0]];
        const float gold = part + __shfl_xor(part, 16, 32);
        if (!hi) sRed[16 + lh] = gold;   // chains 0..15 of this block
    }

    // ---- per-(vgpr r) emission row bases; C-layout tag columns ----
    const float* emrow[8];
#pragma unroll
    for (int r = 0; r < 8; ++r) {
        const int c  = r + 8 * hi;
        const int bv = blk * 16 + c;
        const int b2 = bv >> 3, v2 = bv & 7;
        const int vl = vlab[b2 * VV + v2];
        emrow[r] = score + (size_t)((b2 * SS + vl) * SS) * TT;
    }
    int  tagn[NTI], tagc[NTI];
    bool valid[NTI];
#pragma unroll
    for (int n = 0; n < NTI; ++n) {
        tagn[n]  = n * 16 + lh;
        valid[n] = tagn[n] < TT;
        tagc[n]  = valid[n] ? tagn[n] : (TT - 1);   // clamped (no OOB loads)
    }

    // ---- init: alpha0 = start + emit0; p = exp(alpha0 - m0); logscale = m0 ----
    float pc[NTI][8];     // current normalized p, C layout (chain = r+8*hi, tag col)
    float logscale[8];
    {
        float a0[NTI][8];
#pragma unroll
        for (int n = 0; n < NTI; ++n) {
            const float st = startT[tagc[n]];
#pragma unroll
            for (int r = 0; r < 8; ++r)
                a0[n][r] = valid[n] ? (st + emrow[r][tagc[n]]) : NEG_BIG;
        }
#pragma unroll
        for (int r = 0; r < 8; ++r) {
            float m = a0[0][r];
#pragma unroll
            for (int n = 1; n < NTI; ++n) m = fmaxf(m, a0[n][r]);
            for (int off = 1; off < 16; off <<= 1)
                m = fmaxf(m, __shfl_xor(m, off, 32));
            logscale[r] = m;
#pragma unroll
            for (int n = 0; n < NTI; ++n)
                pc[n][r] = __expf(a0[n][r] - m);   // padded -> 0
        }
#pragma unroll
        for (int n = 0; n < NTI; ++n)
#pragma unroll
            for (int r = 0; r < 8; ++r)
                sP[(r + 8 * hi) * TP + tagn[n]] = pc[n][r];
    }
    __syncthreads();   // sExpT + sP ready

    // running emission row pointers (row t of each chain)
    const float* prow[8];
#pragma unroll
    for (int r = 0; r < 8; ++r) prow[r] = emrow[r] + TT;

    // ---- forward scan: 127 steps ----
    for (int t = 1; t < SS; ++t) {
        // exp(emissions[t]) -- independent of p, overlaps the GEMM below
        float eE[NTI][8];
#pragma unroll
        for (int n = 0; n < NTI; ++n)
#pragma unroll
            for (int r = 0; r < 8; ++r)
                eE[n][r] = __expf(prow[r][tagc[n]]);

        // prefetch next step's rows into cache (speculative; OOB is dropped)
        if (t < SS - 1) {
#pragma unroll
            for (int r = 0; r < 8; ++r)
                __builtin_prefetch(prow[r] + TT + (lane << 3), 0, 1);
        }

        // A fragments: lane = chain row (M=lh both halves), K split {4k+2hi,+1}
        v2f afrag[NKI];
#pragma unroll
        for (int k = 0; k < NKI; ++k) {
            const float* p = &sP[lh * TP + 4 * k + 2 * hi];
            afrag[k].x = p[0];
            afrag[k].y = p[1];
        }

        // GEMM: s[16 x 80] = p[16 x 68] * expT[68 x 80], 2-way split accum
#pragma unroll
        for (int n = 0; n < NTI; ++n) {
            v8f acc0 = 0.0f, acc1 = 0.0f;
#pragma unroll
            for (int k = 0; k < NKI; ++k) {
                v2f bfrag;
                const float* q = &sExpT[(4 * k + 2 * hi) * TP + n * 16 + lh];
                bfrag.x = q[0];
                bfrag.y = q[TP];
                if (k & 1)
                    acc1 = __builtin_amdgcn_wmma_f32_16x16x4_f32(
                        false, afrag[k], false, bfrag, (short)0, acc1, false, false);
                else
                    acc0 = __builtin_amdgcn_wmma_f32_16x16x4_f32(
                        false, afrag[k], false, bfrag, (short)0, acc0, false, false);
            }
#pragma unroll
            for (int r = 0; r < 8; ++r)
                pc[n][r] = (acc0[r] + acc1[r]) * eE[n][r];   // padded cols stay 0
        }

        // per-chain renormalize: rcp(max), track exact scale via -log(rcp)
#pragma unroll
        for (int r = 0; r < 8; ++r) {
            float m = pc[0][r];
#pragma unroll
            for (int n = 1; n < NTI; ++n) m = fmaxf(m, pc[n][r]);
            for (int off = 1; off < 16; off <<= 1)
                m = fmaxf(m, __shfl_xor(m, off, 32));
            const float rcp = __builtin_amdgcn_rcpf(m);
            logscale[r] -= __logf(rcp);        // consistent with the factor applied
#pragma unroll
            for (int n = 0; n < NTI; ++n) pc[n][r] *= rcp;
        }

        // publish normalized p for next step's A fragments
#pragma unroll
        for (int n = 0; n < NTI; ++n)
#pragma unroll
            for (int r = 0; r < 8; ++r)
                sP[(r + 8 * hi) * TP + tagn[n]] = pc[n][r];
        __syncthreads();

#pragma unroll
        for (int r = 0; r < 8; ++r) prow[r] += TT;
    }

    // ---- log Z = logscale + log( sum_j p_j * exp(end_j) ) per chain ----
    float eEnd[NTI];
#pragma unroll
    for (int n = 0; n < NTI; ++n) eEnd[n] = __expf(endT[tagc[n]]);
    float lz[8];
#pragma unroll
    for (int r = 0; r < 8; ++r) {
        float s = 0.0f;
#pragma unroll
        for (int n = 0; n < NTI; ++n) s += pc[n][r] * eEnd[n];  // padded p == 0
        for (int off = 1; off < 16; off <<= 1)
            s += __shfl_xor(s, off, 32);
        lz[r] = logscale[r] + __logf(s);
    }
    if (lh == 0) {
#pragma unroll
        for (int r = 0; r < 8; ++r) sRed[8 * hi + r] = lz[r];
    }
    __syncthreads();

    // ---- nll contribution: (logZ - gold) / (B*V), atomically accumulated ----
    if (lane < 16) {
        const float nll = sRed[lane] - sRed[16 + lane];
        atomicAdd(out, nll * (1.0f / (float)(BB * VV)));
    }
}

extern "C" void kernel_launch(void* const* d_in, const int* in_sizes, int n_in,
                              void* d_out, int out_size, void* d_ws, size_t ws_size,
                              hipStream_t stream) {
    (void)in_sizes; (void)n_in; (void)out_size; (void)d_ws; (void)ws_size;
    const float* score  = (const float*)d_in[0];
    const float* trans  = (const float*)d_in[1];
    const float* startT = (const float*)d_in[2];
    const float* endT   = (const float*)d_in[3];
    const int*   vlab   = (const int*)d_in[4];
    const int*   rlab   = (const int*)d_in[5];
    float* out = (float*)d_out;

    crf_zero<<<1, 32, 0, stream>>>(out);
    crf_fwd<<<16, 32, 0, stream>>>(score, trans, startT, endT, vlab, rlab, out);
}